// SuperchargingBKT_14860586844437
// MI455X (gfx1250) — compile-verified
//
#include <hip/hip_runtime.h>

// ---------------------------------------------------------------------------
// SuperchargingBKT — segmented parallel scan for a 2-state BKT HMM on gfx1250.
//   B=8192 sequences, T=512 steps, K=1000 KCs, P=50000 problems, S=100000.
// Design: 16 segments of 32 steps per sequence; one segment per lane,
// 2 sequences per wave32. Pass1 builds per-segment 2x2 transition products,
// a ds_bpermute Hillis-Steele scan composes them across lanes, pass2 replays
// each segment with the exact reference recurrence. KC logit tables (20KB)
// are staged to LDS with CDNA5 async global->LDS loads.
// ---------------------------------------------------------------------------

#define K_TAB      1000
#define TAB_STRIDE 1024
#define T_LEN      512
#define B_SEQ      8192
#define NSEG       16
#define TSEG       32
#define NCHUNK     (TSEG / 4)
#define BKT_EPS    1e-8f

#if defined(__has_builtin)
#if __has_builtin(__builtin_amdgcn_global_load_async_to_lds_b32) && \
    __has_builtin(__builtin_amdgcn_s_wait_asynccnt)
#define USE_ASYNC_LDS 1
#endif
#endif
#ifndef USE_ASYNC_LDS
#define USE_ASYNC_LDS 0
#endif

#if USE_ASYNC_LDS
typedef __attribute__((address_space(1))) int GI;   // global int
typedef __attribute__((address_space(3))) int LI;   // LDS int
#endif

__device__ __forceinline__ float fast_sigmoid(float x) {
  // v_exp_f32 + v_rcp_f32 (TRANS ops, co-issue with VALU)
  return __builtin_amdgcn_rcpf(1.0f + __expf(-x));
}

__device__ __forceinline__ float lane_read(float v, int src_lane) {
  return __int_as_float(
      __builtin_amdgcn_ds_bpermute(src_lane << 2, __float_as_int(v)));
}

struct StepP {
  float pT, pF, pG, pS, pom, pou;
};

__device__ __forceinline__ StepP get_params(const float* __restrict__ tab,
                                            const float* __restrict__ omega,
                                            const float* __restrict__ sigma,
                                            const float4* __restrict__ theta4,
                                            int ob, int k, int p, int s) {
  float4 th = theta4[s];
  float om = omega[p];
  float sg = sigma[p];
  StepP r;
  r.pT = fast_sigmoid(tab[1 * TAB_STRIDE + k] + th.x);
  r.pF = fast_sigmoid(tab[2 * TAB_STRIDE + k] - th.y);
  r.pG = fast_sigmoid(tab[3 * TAB_STRIDE + k] + om + th.z);
  r.pS = fast_sigmoid(tab[4 * TAB_STRIDE + k] + sg - th.w);
  r.pom = ob ? (1.0f - r.pS) : r.pS;   // P(y | mastered)
  r.pou = ob ? r.pG : (1.0f - r.pG);   // P(y | unmastered)
  return r;
}

__device__ __forceinline__ void unpack4(int4 v, int* a) {
  a[0] = v.x; a[1] = v.y; a[2] = v.z; a[3] = v.w;
}

__global__ __launch_bounds__(256) void SuperchargingBKT_scan_kernel(
    const float* __restrict__ pL0_logit, const float* __restrict__ pT_logit,
    const float* __restrict__ pF_logit, const float* __restrict__ pG_logit,
    const float* __restrict__ pS_logit, const float* __restrict__ omega,
    const float* __restrict__ sigma, const float* __restrict__ student_ability,
    const int* __restrict__ obs, const int* __restrict__ kc,
    const int* __restrict__ pid, const int* __restrict__ sid,
    float* __restrict__ out) {
  __shared__ float tab[5 * TAB_STRIDE];
  const int tid = threadIdx.x;

  // ---- Stage the 5 KC logit tables (20KB) into LDS (async on CDNA5) ----
  {
    const float* srcs[5] = {pL0_logit, pT_logit, pF_logit, pG_logit, pS_logit};
#pragma unroll
    for (int tbl = 0; tbl < 5; ++tbl) {
      const float* s = srcs[tbl];
      float* d = tab + tbl * TAB_STRIDE;
      for (int i = tid; i < K_TAB; i += 256) {
#if USE_ASYNC_LDS
        __builtin_amdgcn_global_load_async_to_lds_b32((GI*)(s + i),
                                                      (LI*)(d + i), 0, 0);
#else
        d[i] = s[i];
#endif
      }
    }
#if USE_ASYNC_LDS
    __builtin_amdgcn_s_wait_asynccnt(0);
#endif
    __syncthreads();
  }

  // ---- Lane -> (sequence, segment) mapping: 2 sequences per wave32 ----
  const int wave = tid >> 5;
  const int lane = tid & 31;
  const int half = (lane >> 4) & 1;
  const int seg = lane & 15;
  const int b = ((((int)blockIdx.x * 8 + wave) << 1) + half);
  const size_t rowBase = (size_t)b * T_LEN;
  const int t0 = seg * TSEG;

  const int4* __restrict__ obs4 = (const int4*)(obs + rowBase + t0);
  const int4* __restrict__ kc4 = (const int4*)(kc + rowBase + t0);
  const int4* __restrict__ pid4 = (const int4*)(pid + rowBase + t0);
  const int4* __restrict__ sid4 = (const int4*)(sid + rowBase + t0);
  const float4* __restrict__ theta4 = (const float4*)student_ability;

  // ---- Initial hidden state from pL0 at the sequence's first KC ----
  const int k0 = kc[rowBase];
  const float pL0 = fast_sigmoid(tab[k0]);
  const float hu0 = 1.0f - pL0;
  const float hm0 = pL0;

  // ---- Pass 1: per-segment 2x2 transition product  M = S_31 ... S_0 ----
  // State vector [hu; hm]:  S = [[(1-pT)*pou, pF*pom], [pT*pou, (1-pF)*pom]]
  float m00 = 1.0f, m01 = 0.0f, m10 = 0.0f, m11 = 1.0f;
  for (int c = 0; c < NCHUNK; ++c) {
    int4 ov = obs4[c], kv = kc4[c], pv = pid4[c], sv = sid4[c];
    if (c + 1 < NCHUNK) {  // gfx1250 global_prefetch_b8 for next chunk
      __builtin_prefetch(obs4 + c + 1, 0, 1);
      __builtin_prefetch(kc4 + c + 1, 0, 1);
      __builtin_prefetch(pid4 + c + 1, 0, 1);
      __builtin_prefetch(sid4 + c + 1, 0, 1);
    }
    int oa[4], ka[4], pa[4], sa[4];
    unpack4(ov, oa); unpack4(kv, ka); unpack4(pv, pa); unpack4(sv, sa);
#pragma unroll
    for (int j = 0; j < 4; ++j) {
      StepP pp = get_params(tab, omega, sigma, theta4, oa[j], ka[j], pa[j], sa[j]);
      const float s00 = (1.0f - pp.pT) * pp.pou;
      const float s01 = pp.pF * pp.pom;
      const float s10 = pp.pT * pp.pou;
      const float s11 = (1.0f - pp.pF) * pp.pom;
      const float n00 = s00 * m00 + s01 * m10;
      const float n01 = s00 * m01 + s01 * m11;
      const float n10 = s10 * m00 + s11 * m10;
      const float n11 = s10 * m01 + s11 * m11;
      m00 = n00; m01 = n01; m10 = n10; m11 = n11;
    }
    // rescale once per chunk so 32 composed steps can't underflow
    const float rz = __builtin_amdgcn_rcpf(m00 + m01 + m10 + m11);
    m00 *= rz; m01 *= rz; m10 *= rz; m11 *= rz;
  }

  // ---- Inclusive scan of matrix products across the 16 segment lanes ----
#pragma unroll
  for (int d = 1; d < NSEG; d <<= 1) {
    const int src = lane - d;  // stays within the 16-lane group when seg >= d
    const float a00 = lane_read(m00, src);
    const float a01 = lane_read(m01, src);
    const float a10 = lane_read(m10, src);
    const float a11 = lane_read(m11, src);
    if (seg >= d) {  // compose: own (later steps) x received (earlier steps)
      const float n00 = m00 * a00 + m01 * a10;
      const float n01 = m00 * a01 + m01 * a11;
      const float n10 = m10 * a00 + m11 * a10;
      const float n11 = m10 * a01 + m11 * a11;
      const float rz = __builtin_amdgcn_rcpf(n00 + n01 + n10 + n11);
      m00 = n00 * rz; m01 = n01 * rz; m10 = n10 * rz; m11 = n11 * rz;
    }
  }

  // ---- Exclusive prefix (shift by one lane) -> incoming state per segment ----
  const float e00 = lane_read(m00, lane - 1);
  const float e01 = lane_read(m01, lane - 1);
  const float e10 = lane_read(m10, lane - 1);
  const float e11 = lane_read(m11, lane - 1);
  float hu, hm;
  if (seg == 0) {
    hu = hu0; hm = hm0;
  } else {
    const float u = e00 * hu0 + e01 * hm0;
    const float m = e10 * hu0 + e11 * hm0;
    const float rz = __builtin_amdgcn_rcpf(u + m + BKT_EPS);
    hu = u * rz; hm = m * rz;
  }

  // ---- Pass 2: replay segment with exact reference recurrence ----
  float4* __restrict__ outv = (float4*)(out + rowBase + t0);
  for (int c = 0; c < NCHUNK; ++c) {
    int4 ov = obs4[c], kv = kc4[c], pv = pid4[c], sv = sid4[c];
    int oa[4], ka[4], pa[4], sa[4];
    unpack4(ov, oa); unpack4(kv, ka); unpack4(pv, pa); unpack4(sv, sa);
    float pcv[4];
#pragma unroll
    for (int j = 0; j < 4; ++j) {
      StepP pp = get_params(tab, omega, sigma, theta4, oa[j], ka[j], pa[j], sa[j]);
      const float au = pp.pou * hu;
      const float am = pp.pom * hm;
      float nm = (1.0f - pp.pF) * am + pp.pT * au;
      float nu = pp.pF * am + (1.0f - pp.pT) * au;
      const float rz = __builtin_amdgcn_rcpf(nm + nu + BKT_EPS);
      nm *= rz; nu *= rz;
      pcv[j] = (1.0f - pp.pS) * nm + pp.pG * nu;
      hu = nu; hm = nm;
    }
    float4 o;
    o.x = pcv[0]; o.y = pcv[1]; o.z = pcv[2]; o.w = pcv[3];
    outv[c] = o;
  }
}

extern "C" void kernel_launch(void* const* d_in, const int* in_sizes, int n_in,
                              void* d_out, int out_size, void* d_ws,
                              size_t ws_size, hipStream_t stream) {
  const float* pL0_logit = (const float*)d_in[0];
  const float* pT_logit = (const float*)d_in[1];
  const float* pF_logit = (const float*)d_in[2];
  const float* pG_logit = (const float*)d_in[3];
  const float* pS_logit = (const float*)d_in[4];
  const float* omega = (const float*)d_in[5];
  const float* sigma = (const float*)d_in[6];
  const float* student_ability = (const float*)d_in[7];
  const int* obs = (const int*)d_in[8];
  const int* kc = (const int*)d_in[9];
  const int* pid = (const int*)d_in[10];
  const int* sid = (const int*)d_in[11];
  float* out = (float*)d_out;

  // 256 threads = 8 wave32s; each wave handles 2 sequences (16 lanes/seq),
  // so one block covers 16 sequences -> 8192/16 = 512 blocks.
  dim3 grid(B_SEQ / 16), block(256);
  SuperchargingBKT_scan_kernel<<<grid, block, 0, stream>>>(
      pL0_logit, pT_logit, pF_logit, pG_logit, pS_logit, omega, sigma,
      student_ability, obs, kc, pid, sid, out);
}